// GHMRankingLoss_51556787421840
// MI455X (gfx1250) — compile-verified
//
#include <hip/hip_runtime.h>
#include <hip/hip_bf16.h>
#include <math.h>

// ---------------------------------------------------------------------------
// GHM margin-ranking loss, two-pass streaming implementation for gfx1250.
//   Pass 1: 10-bin histogram of g = sigmoid(-(x1-x2)*(2t-1))   (LDS + int atomics,
//           4 sub-histograms/wave to cut same-bin ds_add serialization)
//   Tiny  : w[b] = max(count,1)^-0.75
//   Pass 2: sum(loss * w[bin]) with WMMA-based wave reduction, block partials
//   Final : deterministic fixed-order reduce of partials, * 1/N
// Memory-bound: 192 MiB/pass; pass 2 expected to hit the 192 MB L2 (RT loads).
// Sigmoid uses v_exp_f32 + v_rcp_f32 (no IEEE divide chain) to stay under the
// memory roofline on the VALU pipe.
// ---------------------------------------------------------------------------

#define GHM_BINS 10
#define GHM_NBLOCKS 2048
#define GHM_NTHREADS 256
#define GHM_NWAVES (GHM_NTHREADS / 32)
#define GHM_SUBH 4                       // sub-histograms per wave

typedef float v2f __attribute__((ext_vector_type(2)));
typedef float v8f __attribute__((ext_vector_type(8)));

// Full-EXEC wave32 sum reduction using the matrix pipe:
// B(4x16): VGPR0 row0 = lanes0-15 partials, row1 = lanes16-31 partials, rows2-3 = 0.
// A(16x4) = ones  ->  D[m][n] = v_n + v_{n+16}; every lane then holds c_{lane&15},
// finished with 4 xor-shuffles over the 16 distinct values.
__device__ __forceinline__ float wave_reduce_wmma(float v) {
    v2f a; a[0] = 1.0f; a[1] = 1.0f;   // A all-ones
    v2f b; b[0] = v;    b[1] = 0.0f;   // rows 2,3 zero
    v8f c = {};
    c = __builtin_amdgcn_wmma_f32_16x16x4_f32(
        /*neg_a=*/false, a, /*neg_b=*/false, b,
        /*c_mod=*/(short)0, c, /*reuse_a=*/false, /*reuse_b=*/false);
    float s = c[0];                    // lane L holds column-sum c_{L&15}
    s += __shfl_xor(s, 1, 32);
    s += __shfl_xor(s, 2, 32);
    s += __shfl_xor(s, 4, 32);
    s += __shfl_xor(s, 8, 32);
    return s;                          // all lanes: full 32-lane sum
}

__device__ __forceinline__ void ghm_elem(float o1, float o2, float t,
                                         float& loss, int& bin) {
    float diff = o1 - o2;
    loss = fmaxf(0.0f, -t * diff);                 // margin = 0
    float es = 2.0f * t - 1.0f;                    // reference's expected_sign
    float z  = -diff * es;                         // + margin (0)
    // sigmoid(z) = rcp(1 + exp(-z)): v_exp_f32 + v_rcp_f32, no divide chain
    float g  = __builtin_amdgcn_rcpf(1.0f + __expf(-z));
    int b = (int)floorf(g * (float)GHM_BINS);
    bin = b < 0 ? 0 : (b > GHM_BINS - 1 ? GHM_BINS - 1 : b);
}

__global__ void k_init_hist(unsigned int* __restrict__ hist) {
    if (threadIdx.x < GHM_BINS) hist[threadIdx.x] = 0u;
}

__global__ void k_hist(const float* __restrict__ o1, const float* __restrict__ o2,
                       const float* __restrict__ tg, unsigned int* __restrict__ g_hist,
                       int n) {
    // 4 sub-histograms per wave (selected by lane&3): worst-case same-address
    // conflict degree drops from 32 to 8 on hot bins (0 and 9).
    __shared__ unsigned int lh[GHM_NWAVES * GHM_SUBH][16];
    const int tid  = threadIdx.x;
    const int lane = tid & 31;
    const int row  = (tid >> 5) * GHM_SUBH + (lane & (GHM_SUBH - 1));
    for (int i = tid; i < GHM_NWAVES * GHM_SUBH * 16; i += GHM_NTHREADS)
        ((unsigned int*)lh)[i] = 0u;
    __syncthreads();

    const int n4 = n >> 2;
    const float4* O1 = (const float4*)o1;
    const float4* O2 = (const float4*)o2;
    const float4* TG = (const float4*)tg;
    const int stride = gridDim.x * blockDim.x;
    float l; int bn;
    for (int i = blockIdx.x * blockDim.x + tid; i < n4; i += stride) {
        float4 a = O1[i], b = O2[i], t = TG[i];    // global_load_b128 x3
        ghm_elem(a.x, b.x, t.x, l, bn); atomicAdd(&lh[row][bn], 1u);
        ghm_elem(a.y, b.y, t.y, l, bn); atomicAdd(&lh[row][bn], 1u);
        ghm_elem(a.z, b.z, t.z, l, bn); atomicAdd(&lh[row][bn], 1u);
        ghm_elem(a.w, b.w, t.w, l, bn); atomicAdd(&lh[row][bn], 1u);
    }
    // scalar tail (N % 4), handled once by block 0
    const int tail = n & 3;
    if (blockIdx.x == 0 && tid < tail) {
        int i = (n4 << 2) + tid;
        ghm_elem(o1[i], o2[i], tg[i], l, bn);
        atomicAdd(&lh[row][bn], 1u);
    }
    __syncthreads();
    for (int b = tid; b < GHM_BINS; b += GHM_NTHREADS) {
        unsigned int s = 0u;
        for (int r = 0; r < GHM_NWAVES * GHM_SUBH; ++r) s += lh[r][b];
        atomicAdd(&g_hist[b], s);                  // integer atomics: deterministic
    }
}

__global__ void k_weights(const unsigned int* __restrict__ g_hist,
                          float* __restrict__ w) {
    int t = threadIdx.x;
    if (t < GHM_BINS) {
        float c = fmaxf((float)g_hist[t], 1.0f);   // clamp(min=1)
        w[t] = exp2f(-0.75f * log2f(c));           // c^-ALPHA
    }
}

__global__ void k_wsum(const float* __restrict__ o1, const float* __restrict__ o2,
                       const float* __restrict__ tg, const float* __restrict__ w,
                       float* __restrict__ partials, int n) {
    __shared__ float sw[16];
    __shared__ float wsum[GHM_NWAVES];
    const int tid = threadIdx.x;
    if (tid < GHM_BINS)            sw[tid] = w[tid];
    else if (tid < 16)             sw[tid] = 0.0f;
    __syncthreads();

    float acc = 0.0f;
    const int n4 = n >> 2;
    const float4* O1 = (const float4*)o1;
    const float4* O2 = (const float4*)o2;
    const float4* TG = (const float4*)tg;
    const int stride = gridDim.x * blockDim.x;
    float l; int bn;
    for (int i = blockIdx.x * blockDim.x + tid; i < n4; i += stride) {
        float4 a = O1[i], b = O2[i], t = TG[i];
        ghm_elem(a.x, b.x, t.x, l, bn); acc += l * sw[bn];
        ghm_elem(a.y, b.y, t.y, l, bn); acc += l * sw[bn];
        ghm_elem(a.z, b.z, t.z, l, bn); acc += l * sw[bn];
        ghm_elem(a.w, b.w, t.w, l, bn); acc += l * sw[bn];
    }
    const int tail = n & 3;
    if (blockIdx.x == 0 && tid < tail) {
        int i = (n4 << 2) + tid;
        ghm_elem(o1[i], o2[i], tg[i], l, bn);
        acc += l * sw[bn];
    }
    // reconverged here: full EXEC for the matrix-pipe reduction
    float ws = wave_reduce_wmma(acc);
    if ((tid & 31) == 0) wsum[tid >> 5] = ws;
    __syncthreads();
    if (tid == 0) {
        float s = 0.0f;
        for (int i = 0; i < GHM_NWAVES; ++i) s += wsum[i];
        partials[blockIdx.x] = s;                  // fixed slot: deterministic
    }
}

__global__ void k_final(const float* __restrict__ partials, float* __restrict__ out,
                        int nparts, float inv_n) {
    __shared__ float wsum[GHM_NWAVES];
    const int tid = threadIdx.x;
    float acc = 0.0f;
    for (int i = tid; i < nparts; i += GHM_NTHREADS) acc += partials[i];
    float ws = wave_reduce_wmma(acc);
    if ((tid & 31) == 0) wsum[tid >> 5] = ws;
    __syncthreads();
    if (tid == 0) {
        float s = 0.0f;
        for (int i = 0; i < GHM_NWAVES; ++i) s += wsum[i];
        out[0] = s * inv_n;
    }
}

extern "C" void kernel_launch(void* const* d_in, const int* in_sizes, int n_in,
                              void* d_out, int out_size, void* d_ws, size_t ws_size,
                              hipStream_t stream) {
    const float* o1 = (const float*)d_in[0];
    const float* o2 = (const float*)d_in[1];
    const float* tg = (const float*)d_in[2];   // targets are float32 (+1/-1)
    float* out = (float*)d_out;
    const int n = in_sizes[0];

    unsigned char* ws = (unsigned char*)d_ws;
    unsigned int* hist  = (unsigned int*)(ws);        // 10 x u32
    float*        w     = (float*)(ws + 64);          // 10 x f32
    float*        parts = (float*)(ws + 128);         // GHM_NBLOCKS x f32

    int n4 = n >> 2;
    int blocks = (n4 + GHM_NTHREADS - 1) / GHM_NTHREADS;
    if (blocks > GHM_NBLOCKS) blocks = GHM_NBLOCKS;
    if (blocks < 1) blocks = 1;

    k_init_hist<<<1, 32, 0, stream>>>(hist);
    k_hist   <<<blocks, GHM_NTHREADS, 0, stream>>>(o1, o2, tg, hist, n);
    k_weights<<<1, 32, 0, stream>>>(hist, w);
    k_wsum   <<<blocks, GHM_NTHREADS, 0, stream>>>(o1, o2, tg, w, parts, n);
    k_final  <<<1, GHM_NTHREADS, 0, stream>>>(parts, out, blocks, 1.0f / (float)n);
}